// GCN_Dense_Att_506806141239
// MI455X (gfx1250) — compile-verified
//
#include <hip/hip_runtime.h>

#define N_NODES 8192
#define IN_CH   128
#define OUT_CH  16
#define K_HOPS  3
#define CHUNKS_PER_HOP (N_NODES / 4)   // 2048 K=4 chunks per hop

typedef __attribute__((ext_vector_type(2))) float v2f;
typedef __attribute__((ext_vector_type(8))) float v8f;

// ---------------------------------------------------------------------------
// Kernel 1: build att_k * (x @ w + b) directly in WMMA B-fragment order.
// supf layout: [k][chunk][lane][2] floats, where for chunk m0 = 4*chunk:
//   lane = 16*half + c  (c = output channel 0..15, half selects K=2h..2h+1)
//   j in {0,1}:  value = att_k * ( x[m0 + 2*half + j] . w[:,c] + b[c] )
// This matches the B operand of V_WMMA_F32_16X16X4_F32 (VGPR v, half h -> K=v+2h).
// Total: 3 * 2048 * 32 * 2 = 393216 floats = 1.5 MB in d_ws.
// ---------------------------------------------------------------------------
__global__ __launch_bounds__(256) void support_frag_kernel(
    const float* __restrict__ x, const float* __restrict__ w,
    const float* __restrict__ b, const float* __restrict__ att_param,
    float* __restrict__ supf)
{
    const int t = blockIdx.x * 256 + threadIdx.x;   // [0, 393216)

    // softmax over the 3 attention logits (tiny, recomputed per thread)
    float a0 = att_param[0], a1 = att_param[1], a2 = att_param[2];
    float mx = fmaxf(a0, fmaxf(a1, a2));
    float e0 = __expf(a0 - mx), e1 = __expf(a1 - mx), e2 = __expf(a2 - mx);
    float inv = 1.0f / (e0 + e1 + e2);
    float att0 = e0 * inv, att1 = e1 * inv, att2 = e2 * inv;

    const int j     = t & 1;
    const int lane  = (t >> 1) & 31;
    const int chunk = (t >> 6) & (CHUNKS_PER_HOP - 1);
    const int k     = t >> 17;                      // 131072 entries per hop
    const int half  = lane >> 4;
    const int c     = lane & 15;
    const int m     = chunk * 4 + half * 2 + j;

    const float* __restrict__ xr = x + (size_t)m * IN_CH;
    float dot = 0.0f;
#pragma unroll 8
    for (int cc = 0; cc < IN_CH; ++cc)
        dot = fmaf(xr[cc], w[cc * OUT_CH + c], dot);

    float attk = (k == 0) ? att0 : ((k == 1) ? att1 : att2);
    supf[t] = attk * (dot + b[c]);
}

// ---------------------------------------------------------------------------
// Kernel 2: out = concat_k(adj_k) @ concat_k(att_k * support)  via f32 WMMA.
// One block per 16-row output tile (512 blocks). 8 waves per block split the
// 24576-deep K dimension; partials reduced through LDS.
// A fragment (16x4 f32): lane l (l<16): row M=l, K=0..1 ; l>=16: row l-16, K=2..3
//  -> each lane issues one float2 global load per WMMA, walking its adj row
//     contiguously (full 128B line consumed over 8 consecutive chunks).
// ---------------------------------------------------------------------------
__global__ __launch_bounds__(256) void gcn_att_wmma_kernel(
    const float* __restrict__ adj, const float* __restrict__ supf,
    float* __restrict__ out)
{
    __shared__ float red[8 * 256];

    const int tid  = threadIdx.x;
    const int wave = tid >> 5;        // 0..7
    const int lane = tid & 31;
    const int half = lane >> 4;       // 0/1 -> K pair select
    const int lr   = lane & 15;       // A row within tile / B column
    const int blk  = blockIdx.x;      // 0..511 row tiles
    const int row  = blk * 16 + lr;   // adj row this lane streams

    v8f acc = {};

    for (int k = 0; k < K_HOPS; ++k) {
        // wave w covers m in [w*1024, (w+1)*1024) for this hop
        const float* ap = adj
            + ((size_t)k * N_NODES + (size_t)row) * (size_t)N_NODES
            + (size_t)(wave * 1024 + half * 2);
        const float* bp = supf
            + (size_t)(k * CHUNKS_PER_HOP + wave * 256) * 64
            + (size_t)(lane * 2);

        for (int cb = 0; cb < 32; ++cb) {          // 32 blocks of 8 chunks
            __builtin_prefetch(ap + 64, 0, 0);     // global_prefetch_b8, ~2 lines ahead
#pragma unroll
            for (int u = 0; u < 8; ++u) {
                v2f a  = *(const v2f*)ap;  ap += 4;   // 16B of row per chunk (2 lanes)
                v2f bb = *(const v2f*)bp;  bp += 64;  // coalesced 256B per wave
                // D = A(16x4) * B(4x16) + C, all f32
                acc = __builtin_amdgcn_wmma_f32_16x16x4_f32(
                        false, a, false, bb, (short)0, acc, false, false);
            }
        }
    }

    // cross-wave reduction of the 16x16 f32 tile through LDS
#pragma unroll
    for (int vg = 0; vg < 8; ++vg)
        red[wave * 256 + vg * 32 + lane] = acc[vg];
    __syncthreads();

    float s = 0.0f;
#pragma unroll
    for (int ww = 0; ww < 8; ++ww)
        s += red[ww * 256 + tid];

    // C/D layout: VGPR vg, lane ln -> M = vg + 8*(ln>=16), N = ln & 15
    const int vg = tid >> 5;
    const int ln = tid & 31;
    const int M  = vg + ((ln >> 4) << 3);
    const int Nc = ln & 15;
    out[((size_t)blk * 16 + M) * OUT_CH + Nc] = s;
}

// ---------------------------------------------------------------------------
extern "C" void kernel_launch(void* const* d_in, const int* in_sizes, int n_in,
                              void* d_out, int out_size, void* d_ws, size_t ws_size,
                              hipStream_t stream)
{
    const float* x   = (const float*)d_in[0];   // [8192,128]
    const float* w   = (const float*)d_in[1];   // [128,16]
    const float* b   = (const float*)d_in[2];   // [16]
    const float* adj = (const float*)d_in[3];   // [3,8192,8192]
    const float* att = (const float*)d_in[4];   // [3]
    float* out  = (float*)d_out;                // [8192,16]
    float* supf = (float*)d_ws;                 // 1.5 MB fragment-packed B

    // 3*2048*64 = 393216 fragment values -> 1536 blocks of 256
    support_frag_kernel<<<1536, 256, 0, stream>>>(x, w, b, att, supf);
    // 512 output row-tiles, 8 waves each
    gcn_att_wmma_kernel<<<512, 256, 0, stream>>>(adj, supf, out);
}